// WindowAttention_50732153700565
// MI455X (gfx1250) — compile-verified
//
#include <hip/hip_runtime.h>

typedef __attribute__((ext_vector_type(2))) float v2f;
typedef __attribute__((ext_vector_type(8))) float v8f;

#define QP 100                   // q/k/v/out row pitch: 400B = 16B-aligned, bank-conflict-free
#define AP 65                    // attn scratch row pitch (odd)

// LDS layout in floats (all region starts 16B-aligned)
#define Q_OFF   0
#define K_OFF   (Q_OFF + 64*QP)
#define V_OFF   (K_OFF + 64*QP)
#define M_OFF   (V_OFF + 64*QP)          // mask tile 64x64
#define AT_OFF  (M_OFF + 64*64)          // per-wave attn scratch: 8 waves x 16 x AP
#define O_OFF   (AT_OFF + 8*16*AP)       // output tile 64xQP
#define SM_FLOATS (O_OFF + 64*QP)        // 38,016 floats = 152,064 B

// 16-lane XOR butterfly step via DPP16 row_xmask (no LDS, no dscnt stall).
template <int MASK>
__device__ __forceinline__ float dpp_xor16(float v) {
    int r = __builtin_amdgcn_mov_dpp(__float_as_int(v), 0x160 | MASK, 0xF, 0xF, true);
    return __int_as_float(r);
}
template <int MASK>
__device__ __forceinline__ void red_max(float& v) { v = fmaxf(v, dpp_xor16<MASK>(v)); }
template <int MASK>
__device__ __forceinline__ void red_sum(float& v) { v += dpp_xor16<MASK>(v); }

// Async direct global->LDS 16B copy (ASYNCcnt-tracked, no VGPR round trip).
__device__ __forceinline__ void async_b128(unsigned lds_byte, const void* g) {
    asm volatile("global_load_async_to_lds_b128 %0, %1, off"
                 :: "v"(lds_byte), "v"(g) : "memory");
}

__global__ void __launch_bounds__(256)
win_attn_kernel(const float* __restrict__ x,
                const float* __restrict__ mask,
                const float* __restrict__ lepe_w,
                const float* __restrict__ lepe_b,
                float* __restrict__ out,
                int Bsz)
{
    extern __shared__ float sm[];
    const int tid  = threadIdx.x;
    const int blk  = blockIdx.x;
    const int widx = blk & 1023;              // window index inside an image
    const int b    = blk >> 10;
    const int wy   = widx >> 5, wx = widx & 31;

    const size_t tstride = (size_t)Bsz * 256 * 256 * 96;   // one tensor of x
    const size_t wbase   = (((size_t)b * 256 + wy * 8) * 256 + wx * 8) * 96;

    // flat shared pointer truncates to the LDS byte offset (addr[31:0])
    const unsigned smbase = (unsigned)(uintptr_t)(void*)sm;

    // -------- async global->LDS fill of q, k, v and mask (no VGPR staging) ----
    for (int it = 0; it < 6; ++it) {
        int f4  = tid + it * 256;             // 0..1535 float4 slots (64 rows x 24)
        int pos = f4 / 24;
        int c4  = (f4 - pos * 24) * 4;
        int i = pos >> 3, j = pos & 7;
        size_t g = wbase + ((size_t)i * 256 + j) * 96 + c4;
        unsigned row = (unsigned)(pos * QP + c4) * 4u;
        async_b128(smbase + Q_OFF * 4u + row, x + g);
        async_b128(smbase + K_OFF * 4u + row, x + tstride + g);
        async_b128(smbase + V_OFF * 4u + row, x + 2 * tstride + g);
    }
    {
        const float* mrow = mask + (size_t)widx * 4096;
        for (int it = 0; it < 4; ++it) {
            int e4 = tid + it * 256;          // float4 slots over 64x64 mask
            async_b128(smbase + (M_OFF + e4 * 4) * 4u, mrow + e4 * 4);
        }
    }
    asm volatile("s_wait_asynccnt 0" ::: "memory");
    __syncthreads();

    const int wave = tid >> 5;
    const int lane = tid & 31;
    const int half = lane >> 4;               // which 16-lane half of the wave
    const int lm   = lane & 15;

    // ---------------- 24 jobs = (head, 16-row tile) ----------------
    for (int job = wave; job < 24; job += 8) {
        const int h    = job >> 2;
        const int mt   = job & 3;
        const int c0   = h * 16;              // head channel base
        const int row0 = mt * 16;             // query-row base

        // ---- load ALL fragments first so ds loads batch (partial dscnt waits)
        v2f aq[4];
        #pragma unroll
        for (int s = 0; s < 4; ++s) {
            const int kc = c0 + 4 * s + 2 * half;       // K index = vgpr + 2*half
            aq[s].x = sm[Q_OFF + (row0 + lm) * QP + kc];
            aq[s].y = sm[Q_OFF + (row0 + lm) * QP + kc + 1];
        }
        v2f bk[16];
        #pragma unroll
        for (int t = 0; t < 4; ++t)
            #pragma unroll
            for (int s = 0; s < 4; ++s) {
                const int kc = c0 + 4 * s + 2 * half;
                bk[t * 4 + s].x = sm[K_OFF + (t * 16 + lm) * QP + kc];  // B[k][n]=K[n][kc]
                bk[t * 4 + s].y = sm[K_OFF + (t * 16 + lm) * QP + kc + 1];
            }

        // attn(16x64) = Q(16x16) @ K^T(16x64): 4 tiles x 4 k-steps
        v8f acc[4];
        #pragma unroll
        for (int t = 0; t < 4; ++t) {
            v8f c = {};
            #pragma unroll
            for (int s = 0; s < 4; ++s)
                c = __builtin_amdgcn_wmma_f32_16x16x4_f32(false, aq[s], false,
                                                          bk[t * 4 + s],
                                                          (short)0, c, false, false);
            acc[t] = c;
        }

        // scale (1/sqrt(hd)=0.25) + mask add + row-wise softmax
        #pragma unroll
        for (int r = 0; r < 8; ++r) {
            const int grow = row0 + r + 8 * half;
            float mx = -3.0e38f;
            #pragma unroll
            for (int t = 0; t < 4; ++t) {
                float v = fmaf(acc[t][r], 0.25f, sm[M_OFF + grow * 64 + t * 16 + lm]);
                acc[t][r] = v;
                mx = fmaxf(mx, v);
            }
            red_max<1>(mx); red_max<2>(mx); red_max<4>(mx); red_max<8>(mx);
            float ss = 0.f;
            #pragma unroll
            for (int t = 0; t < 4; ++t) {
                float e = __expf(acc[t][r] - mx);
                acc[t][r] = e;
                ss += e;
            }
            red_sum<1>(ss); red_sum<2>(ss); red_sum<4>(ss); red_sum<8>(ss);
            const float inv = 1.0f / ss;
            #pragma unroll
            for (int t = 0; t < 4; ++t) acc[t][r] *= inv;
        }

        // stage softmaxed tile to per-wave LDS scratch (C-layout -> A-layout)
        float* at = &sm[AT_OFF + wave * 16 * AP];
        #pragma unroll
        for (int t = 0; t < 4; ++t)
            #pragma unroll
            for (int r = 0; r < 8; ++r)
                at[(r + 8 * half) * AP + t * 16 + lm] = acc[t][r];
        asm volatile("s_wait_dscnt 0" ::: "memory");   // in-wave LDS RAW

        // ---- batch-load PV fragments, then 16 WMMAs
        v2f ap[16], bv[16];
        #pragma unroll
        for (int kt = 0; kt < 16; ++kt) {
            const int kk = kt * 4 + 2 * half;
            ap[kt].x = at[lm * AP + kk];
            ap[kt].y = at[lm * AP + kk + 1];
            bv[kt].x = sm[V_OFF + kk * QP + c0 + lm];       // B[k][n] = V[k][c0+n]
            bv[kt].y = sm[V_OFF + (kk + 1) * QP + c0 + lm];
        }
        v8f o = {};
        #pragma unroll
        for (int kt = 0; kt < 16; ++kt)
            o = __builtin_amdgcn_wmma_f32_16x16x4_f32(false, ap[kt], false, bv[kt],
                                                      (short)0, o, false, false);
        #pragma unroll
        for (int r = 0; r < 8; ++r)
            sm[O_OFF + (row0 + r + 8 * half) * QP + c0 + lm] = o[r];
    }
    __syncthreads();

    // ---------------- LePE depthwise 3x3 (within window) + fused store ------
    for (int it = 0; it < 24; ++it) {
        int e   = tid + it * 256;             // 0..6143 over (pos, c)
        int pos = e / 96;
        int c   = e - pos * 96;
        int i = pos >> 3, j = pos & 7;
        float accv = lepe_b[c];
        const float* wch = lepe_w + c * 9;
        #pragma unroll
        for (int di = -1; di <= 1; ++di) {
            int ii = i + di;
            if (ii < 0 || ii > 7) continue;
            #pragma unroll
            for (int dj = -1; dj <= 1; ++dj) {
                int jj = j + dj;
                if (jj < 0 || jj > 7) continue;
                accv += wch[(di + 1) * 3 + (dj + 1)] * sm[V_OFF + (ii * 8 + jj) * QP + c];
            }
        }
        float val = sm[O_OFF + pos * QP + c] + accv;
        out[wbase + ((size_t)i * 256 + j) * 96 + c] = val;
    }
}

extern "C" void kernel_launch(void* const* d_in, const int* in_sizes, int n_in,
                              void* d_out, int out_size, void* d_ws, size_t ws_size,
                              hipStream_t stream) {
    const float* x    = (const float*)d_in[0];
    const float* mask = (const float*)d_in[1];
    const float* lw   = (const float*)d_in[2];
    const float* lb   = (const float*)d_in[3];
    float* out = (float*)d_out;

    int Bsz = in_sizes[0] / (3 * 256 * 256 * 96);   // = 8 for the reference shapes
    int nblocks = Bsz * 32 * 32;                    // one block per 8x8 window
    size_t shmem = (size_t)SM_FLOATS * sizeof(float);
    win_attn_kernel<<<nblocks, 256, shmem, stream>>>(x, mask, lw, lb, out, Bsz);
}